// MoEFeedForward_78340203479522
// MI455X (gfx1250) — compile-verified
//
#include <hip/hip_runtime.h>
#include <math.h>

// ---------------- problem constants (from reference) ----------------
#define D_MODEL 1024
#define D_FF    4096
#define N_EXP   8
#define T_TOK   4096      // 2 * 2048 tokens
#define TOPK    2

// ---------------- tiling ----------------
#define M_TILE  32        // tokens per FFN block
#define FC      64        // ff-chunk width
#define XS_LD   1028      // padded stride: 16B-aligned rows, banks = 4*row+col
#define WS_LD   68        // padded stride for 64-wide staging tiles
#define HS_LD   68
#define WS_SZ   (64 * WS_LD)   // one staging buffer (floats)

typedef __attribute__((ext_vector_type(2))) float v2f;
typedef __attribute__((ext_vector_type(8))) float v8f;
typedef int vi4 __attribute__((vector_size(16)));   // async builtin's pointee type

// ---------------- CDNA5 async global->LDS staging --------------------
#if defined(__gfx1250__) && __has_builtin(__builtin_amdgcn_global_load_async_to_lds_b128)
#define HAVE_ASYNC_LDS 1
#else
#define HAVE_ASYNC_LDS 0
#endif

__device__ __forceinline__ void async_copy16(const float* g, float* l) {
#if HAVE_ASYNC_LDS
  // per-lane 16B global->LDS, no VGPR data, tracked by ASYNCcnt
  __builtin_amdgcn_global_load_async_to_lds_b128(
      (vi4*)(uintptr_t)g, (vi4*)l, 0, 0);
#else
  const float4 v = *(const float4*)g;
  l[0] = v.x; l[1] = v.y; l[2] = v.z; l[3] = v.w;
#endif
}

__device__ __forceinline__ void async_wait0() {
#if HAVE_ASYNC_LDS
#if __has_builtin(__builtin_amdgcn_s_wait_asynccnt)
  __builtin_amdgcn_s_wait_asynccnt(0);
#else
  asm volatile("s_wait_asynccnt 0x0" ::: "memory");
#endif
#endif
}

// stage a 64x64 f32 tile (global row stride ldg) into an LDS buffer
__device__ __forceinline__
void stage64x64(const float* __restrict__ gbase, size_t ldg,
                float* __restrict__ lbuf, int tid) {
#pragma unroll
  for (int p = 0; p < 4; ++p) {
    const int r = (tid >> 4) + (p << 4);     // 0..63
    const int c = (tid & 15) << 2;           // 0..60, float4 granularity
    async_copy16(gbase + (size_t)r * ldg + c, lbuf + r * WS_LD + c);
  }
}

__device__ __forceinline__ v2f ldfrag(const float* p) {
  v2f r; r.x = p[0]; r.y = p[1]; return r;
}

// =====================================================================
// init: zero output y region and the per-expert cursors/counts
// =====================================================================
__global__ void moe_init(float* __restrict__ y, int* __restrict__ counts, int n) {
  int i = blockIdx.x * blockDim.x + threadIdx.x;
  if (i < 16) counts[i] = 0;
  for (int j = i; j < n; j += gridDim.x * blockDim.x) y[j] = 0.0f;
}

// =====================================================================
// router: one wave32 per token.  logits = x[t] . Wg[e], top-2, softmax,
// append token into per-expert compacted list.  counts[e] doubles as the
// usage count for the aux loss.
// =====================================================================
__global__ void moe_router(const float* __restrict__ x,
                           const float* __restrict__ Wg,
                           int*   __restrict__ counts,
                           int*   __restrict__ list,
                           float* __restrict__ plist) {
  const int lane = threadIdx.x & 31;
  const int t    = blockIdx.x * (blockDim.x >> 5) + (threadIdx.x >> 5);
  if (t >= T_TOK) return;

  const float* xr = x + (size_t)t * D_MODEL;
  float acc[N_EXP];
#pragma unroll
  for (int e = 0; e < N_EXP; ++e) acc[e] = 0.0f;

  for (int d = lane; d < D_MODEL; d += 32) {
    const float xv = xr[d];
#pragma unroll
    for (int e = 0; e < N_EXP; ++e) acc[e] += xv * Wg[e * D_MODEL + d];
  }
#pragma unroll
  for (int off = 16; off > 0; off >>= 1) {
#pragma unroll
    for (int e = 0; e < N_EXP; ++e) acc[e] += __shfl_xor(acc[e], off, 32);
  }

  if (lane == 0) {
    int i0 = 0; float s0 = acc[0];
#pragma unroll
    for (int e = 1; e < N_EXP; ++e) if (acc[e] > s0) { s0 = acc[e]; i0 = e; }
    int i1 = -1; float s1 = -3.4e38f;
#pragma unroll
    for (int e = 0; e < N_EXP; ++e)
      if (e != i0 && acc[e] > s1) { s1 = acc[e]; i1 = e; }
    const float p0 = 1.0f / (1.0f + __expf(s1 - s0));
    const float p1 = 1.0f - p0;

    int pos0 = atomicAdd(&counts[i0], 1);
    list [i0 * T_TOK + pos0] = t;
    plist[i0 * T_TOK + pos0] = p0;
    int pos1 = atomicAdd(&counts[i1], 1);
    list [i1 * T_TOK + pos1] = t;
    plist[i1 * T_TOK + pos1] = p1;
  }
}

// =====================================================================
// fused expert FFN, per block = (expert e, 32 gathered tokens).
//   LDS: Xs 32x1024, Ys 32x1024 accumulator, H chunk 32x64,
//        double-buffered 64x64 staging tile fed by async global->LDS,
//        staging always one 16-wmma step ahead (incl. phase/chunk edges).
//   per ff-chunk: H = relu(Xs @ W1_chunk^T)  (K=1024)
//                 Ys += H @ W2_slice^T       (K=64, 16 n-blocks)
//   all MACs via v_wmma_f32_16x16x4_f32, depth-4 pipelined LDS frag loads.
// =====================================================================
__global__ __launch_bounds__(256, 1)
void moe_ffn(const float* __restrict__ x,
             const float* __restrict__ W1,
             const float* __restrict__ W2,
             const int*   __restrict__ counts,
             const int*   __restrict__ list,
             const float* __restrict__ plist,
             float* __restrict__ y) {
  extern __shared__ float smem[];
  float* Xs = smem;                      // 32 * XS_LD
  float* Ys = Xs + M_TILE * XS_LD;       // 32 * XS_LD
  float* Hs = Ys + M_TILE * XS_LD;       // 32 * HS_LD
  float* Ws = Hs + M_TILE * HS_LD;       // 2 * 64 * WS_LD (double buffer)
  __shared__ int   toks[M_TILE];
  __shared__ float prs [M_TILE];

  const int e   = blockIdx.y;
  const int cnt = counts[e];
  const int m0  = blockIdx.x * M_TILE;
  if (m0 >= cnt) return;

  const int tid = threadIdx.x;
  if (tid < M_TILE) {
    const int i = m0 + tid;
    const bool v = (i < cnt);
    toks[tid] = v ? list [e * T_TOK + i] : 0;     // pad rows read token 0,
    prs [tid] = v ? plist[e * T_TOK + i] : 0.0f;  // contribute with weight 0
  }
  __syncthreads();

  // gathered X rows -> LDS (async b128 copies), Ys zeroed alongside
  for (int i = tid; i < M_TILE * (D_MODEL / 4); i += 256) {
    const int r = i >> 8;                // D_MODEL/4 == 256
    const int c = (i & 255) << 2;
    async_copy16(x + (size_t)toks[r] * D_MODEL + c, Xs + r * XS_LD + c);
  }
  for (int i = tid; i < M_TILE * XS_LD; i += 256) Ys[i] = 0.0f;

  const int wave = tid >> 5;
  const int lane = tid & 31;
  const int half = lane >> 4;            // wmma lane-half
  const int lid  = lane & 15;
  const int koff = half << 1;            // k-offset of this lane-half
  const int h_m  = (wave & 1) << 4;      // H-tile: 2 m-subtiles x 4 n-subtiles
  const int h_n  = (wave >> 1) << 4;
  const int ym   = (wave & 1) << 4;      // Y-tile ownership within an n-block
  const int nloc = (wave >> 1) << 4;

  const float* W1e = W1 + (size_t)e * D_FF * D_MODEL;
  const float* W2e = W2 + (size_t)e * D_MODEL * D_FF;

  // cold stage: chunk 0, W1 K-slab 0 -> buf0 (everything after is prefetched)
  stage64x64(W1e, D_MODEL, Ws, tid);
  async_wait0();
  __syncthreads();

  for (int f0 = 0; f0 < D_FF; f0 += FC) {
    // -------- phase 1: H = relu(X @ W1_chunk^T), K = 1024 ---------------
    v8f acc;
#pragma unroll
    for (int j = 0; j < 8; ++j) acc[j] = 0.0f;

    const float* arow1 = Xs + (h_m + lid) * XS_LD;
#pragma unroll 1
    for (int s = 0; s < 16; ++s) {
      const int d0 = s << 6;
      const float* cur  = Ws + (s & 1) * WS_SZ;
      float*       nxt  = Ws + ((s + 1) & 1) * WS_SZ;
      if (s < 15)        // prefetch next K-slab of W1
        stage64x64(W1e + (size_t)f0 * D_MODEL + d0 + 64, D_MODEL, nxt, tid);
      else               // prefetch phase-2 first W2 n-block
        stage64x64(W2e + f0, D_FF, nxt, tid);

      const float* ap = arow1 + d0 + koff;
      const float* bp = cur + (h_n + lid) * WS_LD + koff;
      v2f A[4], B[4];
#pragma unroll
      for (int q = 0; q < 4; ++q) { A[q] = ldfrag(ap + (q << 2)); B[q] = ldfrag(bp + (q << 2)); }
#pragma unroll
      for (int q = 0; q < 16; ++q) {
        acc = __builtin_amdgcn_wmma_f32_16x16x4_f32(
                  false, A[q & 3], false, B[q & 3], (short)0, acc, false, false);
        if (q + 4 < 16) {
          A[q & 3] = ldfrag(ap + ((q + 4) << 2));
          B[q & 3] = ldfrag(bp + ((q + 4) << 2));
        }
      }
      async_wait0();
      __syncthreads();
    }
    // relu -> Hs  (C/D layout: vgpr j holds rows j / j+8)
#pragma unroll
    for (int j = 0; j < 8; ++j)
      Hs[(h_m + j + (half << 3)) * HS_LD + h_n + lid] = fmaxf(acc[j], 0.0f);
    __syncthreads();

    // -------- phase 2: Ys += H @ W2_slice^T, K = 64, 16 n-blocks --------
    const float* arow2 = Hs + (ym + lid) * HS_LD;
#pragma unroll 1
    for (int nb = 0; nb < 16; ++nb) {
      const float* cur = Ws + (nb & 1) * WS_SZ;
      float*       nxt = Ws + ((nb + 1) & 1) * WS_SZ;
      if (nb < 15)       // prefetch next W2 n-block
        stage64x64(W2e + (size_t)(nb + 1) * 64 * D_FF + f0, D_FF, nxt, tid);
      else if (f0 + FC < D_FF)   // prefetch next chunk's first W1 K-slab
        stage64x64(W1e + (size_t)(f0 + FC) * D_MODEL, D_MODEL, nxt, tid);

      const int yn = (nb << 6) + nloc;
      v8f cacc;
#pragma unroll
      for (int j = 0; j < 8; ++j)
        cacc[j] = Ys[(ym + j + (half << 3)) * XS_LD + yn + lid];

      const float* ap = arow2 + koff;
      const float* bp = cur + (nloc + lid) * WS_LD + koff;
      v2f A[4], B[4];
#pragma unroll
      for (int q = 0; q < 4; ++q) { A[q] = ldfrag(ap + (q << 2)); B[q] = ldfrag(bp + (q << 2)); }
#pragma unroll
      for (int q = 0; q < 16; ++q) {
        cacc = __builtin_amdgcn_wmma_f32_16x16x4_f32(
                   false, A[q & 3], false, B[q & 3], (short)0, cacc, false, false);
        if (q + 4 < 16) {
          A[q & 3] = ldfrag(ap + ((q + 4) << 2));
          B[q & 3] = ldfrag(bp + ((q + 4) << 2));
        }
      }
#pragma unroll
      for (int j = 0; j < 8; ++j)
        Ys[(ym + j + (half << 3)) * XS_LD + yn + lid] = cacc[j];
      async_wait0();
      __syncthreads();
    }
  }

  // gating-weighted scatter-add into y (2 contributions per token total)
  for (int i = tid; i < M_TILE * D_MODEL; i += 256) {
    const int r = i >> 10, col = i & (D_MODEL - 1);
    const float v = prs[r] * Ys[r * XS_LD + col];
    unsafeAtomicAdd(y + (size_t)toks[r] * D_MODEL + col, v);
  }
}

// =====================================================================
// aux loss: usage[e] = counts[e]/T; ema = 0.01*usage;
//           p = ema/(sum+1e-9); aux = sum(p^2)*E
// =====================================================================
__global__ void moe_aux(const int* __restrict__ counts, float* __restrict__ out_aux) {
  const int lane = threadIdx.x & 31;
  float ema = (lane < N_EXP) ? 0.01f * ((float)counts[lane] / (float)T_TOK) : 0.0f;
  float s = ema;
#pragma unroll
  for (int off = 16; off > 0; off >>= 1) s += __shfl_xor(s, off, 32);
  const float p = ema / (s + 1e-9f);
  float v = p * p;
#pragma unroll
  for (int off = 16; off > 0; off >>= 1) v += __shfl_xor(v, off, 32);
  if (lane == 0) out_aux[0] = v * (float)N_EXP;
}

// =====================================================================
extern "C" void kernel_launch(void* const* d_in, const int* in_sizes, int n_in,
                              void* d_out, int out_size, void* d_ws, size_t ws_size,
                              hipStream_t stream) {
  const float* x  = (const float*)d_in[0];
  const float* Wg = (const float*)d_in[1];
  const float* W1 = (const float*)d_in[2];
  const float* W2 = (const float*)d_in[3];
  // d_in[4] = k (== 2, baked into TOPK)

  float* y = (float*)d_out;                          // [T, D] then aux scalar

  int*   counts = (int*)d_ws;
  int*   list   = (int*)((char*)d_ws + 256);
  float* plist  = (float*)((char*)d_ws + 256 + (size_t)N_EXP * T_TOK * 4);

  moe_init<<<2048, 256, 0, stream>>>(y, counts, T_TOK * D_MODEL);
  moe_router<<<T_TOK / 8, 256, 0, stream>>>(x, Wg, counts, list, plist);

  const size_t ldsBytes =
      (size_t)(2 * M_TILE * XS_LD + M_TILE * HS_LD + 2 * WS_SZ) * sizeof(float);
  dim3 grid(T_TOK / M_TILE, N_EXP);                  // worst-case tiles; empty exit fast
  moe_ffn<<<grid, 256, ldsBytes, stream>>>(x, W1, W2, counts, list, plist, y);

  moe_aux<<<1, 32, 0, stream>>>(counts, y + (size_t)T_TOK * D_MODEL);
}